// MultiHeadAttentionLayer_52450140619078
// MI455X (gfx1250) — compile-verified
//
#include <hip/hip_runtime.h>
#include <hip/hip_bf16.h>

// ---------------------------------------------------------------------------
// MI455X (gfx1250) fused multi-head edge attention
//   N=50000, E=800000, D_IN=128, H=8, D_HEAD=16, H*D_HEAD=128
// Phase 1: Q/K/V = h @ W + b  via v_wmma_f32_16x16x4_f32 (FP32 WMMA, wave32),
//          h tile staged to LDS via GLOBAL_LOAD_ASYNC_TO_LDS_B128 (ASYNCcnt)
// Phase 2: per-edge scores + atomic segment-sum (QKV+out ~104MB < 192MB L2)
// Phase 3: normalize by z
// ---------------------------------------------------------------------------

typedef __attribute__((ext_vector_type(2))) float v2f;
typedef __attribute__((ext_vector_type(8))) float v8f;
typedef __attribute__((ext_vector_type(4))) int   v4i;

#define D_MODEL 128   // D_IN == H*D_HEAD == 128
#define NHEAD   8
#define DHEAD   16
#define LDS_PAD 132   // 128 + 4 -> column reads hit 16 distinct banks

#if defined(__has_builtin)
#if __has_builtin(__builtin_amdgcn_global_load_async_to_lds_b128)
#define HAVE_ASYNC_LDS 1
#endif
#endif

#ifdef HAVE_ASYNC_LDS
// Builtin signature (from hipcc diagnostic): (v4i AS1*, v4i AS3*, imm, imm)
typedef __attribute__((address_space(1))) v4i GV4;
typedef __attribute__((address_space(3))) v4i LV4;
#endif

// ---------------------------------------------------------------------------
// Phase 1: fused QKV projection.
// Block = 256 threads = 8 waves. Block handles 16 output rows.
// Wave w handles columns [16w, 16w+16) for all three outputs, reusing the
// A-fragment across Q/K/V accumulators. K-loop: 32 steps of K=4 FP32 WMMA.
// ---------------------------------------------------------------------------
__global__ __launch_bounds__(256) void qkv_wmma_kernel(
    const float* __restrict__ h,
    const float* __restrict__ Wq, const float* __restrict__ bq,
    const float* __restrict__ Wk, const float* __restrict__ bk,
    const float* __restrict__ Wv, const float* __restrict__ bv,
    float* __restrict__ Q, float* __restrict__ K, float* __restrict__ V,
    int n)
{
    __shared__ float lds_h[16 * LDS_PAD];

    const int tid     = threadIdx.x;
    const int wave    = tid >> 5;
    const int lane    = tid & 31;
    const int laneN   = lane & 15;   // M index for A frag, N index for B/C frags
    const int half    = lane >> 4;   // 0: K=0,1  1: K=2,3 (per ISA 16x4 f32 layout)
    const int rowbase = blockIdx.x * 16;
    const int colbase = wave * 16;

    // Stage the 16x128 h tile into LDS. Preferred path: async global->LDS DMA
    // (no VGPR round trip, tracked by ASYNCcnt). Fallback: b128 load + ds_store.
    {
        const float* gsrc = h + (size_t)rowbase * D_MODEL;
#ifdef HAVE_ASYNC_LDS
        for (int i = tid; i < 16 * (D_MODEL / 4); i += 256) {
            int r = i >> 5;           // row within tile (32 float4 per row)
            int c = i & 31;           // float4 column
            float* lp = &lds_h[r * LDS_PAD + c * 4];   // 16B chunk stays in-row
            if (rowbase + r < n) {
                __builtin_amdgcn_global_load_async_to_lds_b128(
                    (GV4*)(gsrc + (size_t)i * 4), (LV4*)lp, 0, 0);
            } else {
                lp[0] = 0.0f; lp[1] = 0.0f; lp[2] = 0.0f; lp[3] = 0.0f;
            }
        }
#if __has_builtin(__builtin_amdgcn_s_wait_asynccnt)
        __builtin_amdgcn_s_wait_asynccnt(0);
#else
        asm volatile("s_wait_asynccnt 0" ::: "memory");
#endif
#else
        const float4* gsrc4 = (const float4*)gsrc;
        for (int i = tid; i < 16 * (D_MODEL / 4); i += 256) {
            int r = i >> 5;
            int c = i & 31;
            float4 vv;
            if (rowbase + r < n) vv = gsrc4[i];
            else { vv.x = vv.y = vv.z = vv.w = 0.0f; }
            float* drow = &lds_h[r * LDS_PAD + c * 4];
            drow[0] = vv.x; drow[1] = vv.y; drow[2] = vv.z; drow[3] = vv.w;
        }
#endif
    }
    __syncthreads();

    v8f accQ = {}; v8f accK = {}; v8f accV = {};

    const float* hrow = &lds_h[laneN * LDS_PAD];

    #pragma unroll 4
    for (int kb = 0; kb < D_MODEL; kb += 4) {
        const int k0 = kb + 2 * half;              // this lane holds K=k0, k0+1
        v2f a;
        a.x = hrow[k0];
        a.y = hrow[k0 + 1];

        const int woff0 = k0 * D_MODEL + colbase + laneN;
        v2f bq2; bq2.x = Wq[woff0]; bq2.y = Wq[woff0 + D_MODEL];
        v2f bk2; bk2.x = Wk[woff0]; bk2.y = Wk[woff0 + D_MODEL];
        v2f bv2; bv2.x = Wv[woff0]; bv2.y = Wv[woff0 + D_MODEL];

        accQ = __builtin_amdgcn_wmma_f32_16x16x4_f32(false, a, false, bq2,
                                                     (short)0, accQ, false, false);
        accK = __builtin_amdgcn_wmma_f32_16x16x4_f32(false, a, false, bk2,
                                                     (short)0, accK, false, false);
        accV = __builtin_amdgcn_wmma_f32_16x16x4_f32(false, a, false, bv2,
                                                     (short)0, accV, false, false);
    }

    const float biasq = bq[colbase + laneN];
    const float biask = bk[colbase + laneN];
    const float biasv = bv[colbase + laneN];

    // C/D layout: VGPR r -> M = r + 8*half, N = laneN.
    #pragma unroll
    for (int r = 0; r < 8; ++r) {
        const int row = rowbase + r + 8 * half;
        if (row < n) {
            const size_t off = (size_t)row * D_MODEL + colbase + laneN;
            Q[off] = accQ[r] + biasq;
            K[off] = accK[r] + biask;
            V[off] = accV[r] + biasv;
        }
    }
}

// ---------------------------------------------------------------------------
// Phase 2: per-edge attention + scatter.
// 16 lanes own one (edge, head): coalesced 64B gathers of K[src,h,:], Q[dst,h,:],
// butterfly reduction for the dot product, then one f32 atomic per lane into
// the wV accumulator (= d_out) and one z atomic per group. QKV + out + z are
// L2-resident on MI455X (104 MB < 192 MB), so gathers/atomics stay on-die.
// ---------------------------------------------------------------------------
__global__ __launch_bounds__(256) void edge_attn_kernel(
    const float* __restrict__ Q, const float* __restrict__ K,
    const float* __restrict__ V,
    const int* __restrict__ src, const int* __restrict__ dst,
    float* __restrict__ wV, float* __restrict__ z,
    int npairs /* = E*H */)
{
    const int gid  = blockIdx.x * 256 + threadIdx.x;
    const int pair = gid >> 4;
    if (pair >= npairs) return;
    const int d  = gid & 15;
    const int e  = pair >> 3;   // H == 8
    const int hh = pair & 7;

    const int s = src[e];
    const int t = dst[e];

    const size_t koff = (size_t)s * D_MODEL + hh * DHEAD + d;
    const size_t qoff = (size_t)t * D_MODEL + hh * DHEAD + d;

    float p = K[koff] * Q[qoff];

    // Reduce across the 16-lane group (xor 1,2,4,8 stays inside the group, wave32).
    #pragma unroll
    for (int m = 1; m < 16; m <<= 1)
        p += __shfl_xor(p, m, 32);

    // score = exp(clip(p / sqrt(DHEAD), -5, 5)); sqrt(16) = 4
    float sarg  = fminf(fmaxf(p * 0.25f, -5.0f), 5.0f);
    float score = __expf(sarg);

    const float vv = V[koff];   // V[src] shares src offset with K
    atomicAdd(&wV[qoff], vv * score);
    if (d == 0)
        atomicAdd(&z[(size_t)t * NHEAD + hh], score);
}

// ---------------------------------------------------------------------------
// Phase 3: out = wV / (z + 1e-6), in place on d_out.
// ---------------------------------------------------------------------------
__global__ __launch_bounds__(256) void normalize_kernel(
    float* __restrict__ out, const float* __restrict__ z, int total /* N*128 */)
{
    const int i = blockIdx.x * 256 + threadIdx.x;
    if (i >= total) return;
    const int node = i >> 7;          // / 128
    const int hh   = (i >> 4) & 7;    // head
    out[i] = out[i] / (z[node * NHEAD + hh] + 1e-6f);
}

// ---------------------------------------------------------------------------
extern "C" void kernel_launch(void* const* d_in, const int* in_sizes, int n_in,
                              void* d_out, int out_size, void* d_ws, size_t ws_size,
                              hipStream_t stream)
{
    const float* h  = (const float*)d_in[0];
    const float* Wq = (const float*)d_in[1];
    const float* bq = (const float*)d_in[2];
    const float* Wk = (const float*)d_in[3];
    const float* bk = (const float*)d_in[4];
    const float* Wv = (const float*)d_in[5];
    const float* bv = (const float*)d_in[6];
    const int*   src = (const int*)d_in[7];
    const int*   dst = (const int*)d_in[8];

    const int N = in_sizes[0] / D_MODEL;
    const int E = in_sizes[7];

    float* Q = (float*)d_ws;
    float* K = Q + (size_t)N * D_MODEL;
    float* V = K + (size_t)N * D_MODEL;
    float* z = V + (size_t)N * D_MODEL;
    float* wV = (float*)d_out;   // accumulate wV directly into the output buffer

    // Zero the atomic accumulators every call (graph-capture-safe async memset).
    (void)hipMemsetAsync(wV, 0, (size_t)N * D_MODEL * sizeof(float), stream);
    (void)hipMemsetAsync(z,  0, (size_t)N * NHEAD   * sizeof(float), stream);

    // Phase 1: fused QKV projection (WMMA). N = 50000 = 3125 * 16.
    const int row_tiles = (N + 15) / 16;
    qkv_wmma_kernel<<<row_tiles, 256, 0, stream>>>(h, Wq, bq, Wk, bk, Wv, bv,
                                                   Q, K, V, N);

    // Phase 2: edge scoring + aggregation. 16 threads per (edge, head).
    const long long nthreads = (long long)E * NHEAD * 16;
    const int eblocks = (int)((nthreads + 255) / 256);
    edge_attn_kernel<<<eblocks, 256, 0, stream>>>(Q, K, V, src, dst, wV, z,
                                                  E * NHEAD);

    // Phase 3: normalize.
    const int total = N * D_MODEL;
    normalize_kernel<<<(total + 255) / 256, 256, 0, stream>>>(wV, z, total);
}